// Adam_43757126811972
// MI455X (gfx1250) — compile-verified
//
#include <hip/hip_runtime.h>
#include <math.h>

// Adam "patch" kernel for MI455X (gfx1250).
// HBM-bound streaming op (~465 MB @ 23.3 TB/s => ~20us floor).
// p/g/m/v tiles staged into LDS via CDNA5 async data path
// (global_load_async_to_lds_b128 th:TH_LOAD_NT, ASYNCcnt, s_wait_asynccnt),
// double-buffered; non-temporal b128 stores; transcendental scalar prep
// hoisted into a one-thread setup kernel -> d_ws; inner-loop math uses
// hardware v_sqrt_f32 / v_rcp_f32 (~1 ulp) to keep VALU << HBM time.

typedef float v4f __attribute__((ext_vector_type(4)));

#define TILE_V4 256   // float4s per tile  (= 1024 floats per array)
#define THREADS 256   // 8 wave32s

// ---------------------------------------------------------------- setup ----
// ws[0]=beta1 ws[1]=1-beta1 ws[2]=beta2 ws[3]=1-beta2
// ws[4]=alpha/(1-beta1^t)  ws[5]=1/(1-beta2^t)  ws[6]=eps
__global__ void adam_setup(const float* __restrict__ s_alpha,
                           const float* __restrict__ s_b1,
                           const float* __restrict__ s_b2,
                           const float* __restrict__ s_leps,
                           const int*   __restrict__ s_t,
                           float* __restrict__ ws)
{
    if (threadIdx.x == 0 && blockIdx.x == 0) {
        const float alpha = *s_alpha;
        const float beta1 = 0.5f * (tanhf(*s_b1) + 1.0f);
        const float beta2 = 0.5f * (tanhf(*s_b2) + 1.0f);
        const float eps   = powf(10.0f, *s_leps);
        const float tt    = (float)(*s_t);
        const float bc1   = 1.0f - powf(beta1, tt);
        const float bc2   = 1.0f - powf(beta2, tt);
        ws[0] = beta1;
        ws[1] = 1.0f - beta1;
        ws[2] = beta2;
        ws[3] = 1.0f - beta2;
        ws[4] = alpha / bc1;   // folds m-hat bias correction into the step size
        ws[5] = 1.0f / bc2;    // v-hat bias correction as a multiply
        ws[6] = eps;
    }
}

// ------------------------------------------------------------ main loop ----
__device__ __forceinline__ void async_tile_load(
    v4f (*__restrict__ dst)[TILE_V4],
    const v4f* __restrict__ p4, const v4f* __restrict__ g4,
    const v4f* __restrict__ m4, const v4f* __restrict__ w4,
    long base, int tid)
{
    // LDS byte address = low 32 bits of the generic address of a __shared__ object
    unsigned lp = (unsigned)(size_t)&dst[0][tid];
    unsigned lg = (unsigned)(size_t)&dst[1][tid];
    unsigned lm = (unsigned)(size_t)&dst[2][tid];
    unsigned lv = (unsigned)(size_t)&dst[3][tid];
    const v4f* ap = p4 + base + tid;
    const v4f* ag = g4 + base + tid;
    const v4f* am = m4 + base + tid;
    const v4f* av = w4 + base + tid;
    // VDST = LDS address VGPR, VADDR = 64-bit global address (GV mode).
    // Read-once stream: non-temporal so 465 MB doesn't churn the 192 MB L2.
    asm volatile("global_load_async_to_lds_b128 %0, %1, off th:TH_LOAD_NT" :: "v"(lp), "v"(ap) : "memory");
    asm volatile("global_load_async_to_lds_b128 %0, %1, off th:TH_LOAD_NT" :: "v"(lg), "v"(ag) : "memory");
    asm volatile("global_load_async_to_lds_b128 %0, %1, off th:TH_LOAD_NT" :: "v"(lm), "v"(am) : "memory");
    asm volatile("global_load_async_to_lds_b128 %0, %1, off th:TH_LOAD_NT" :: "v"(lv), "v"(av) : "memory");
}

__device__ __forceinline__ float adam1(float p, float g, float m, float v,
                                       float beta1, float om1,
                                       float beta2, float om2,
                                       float scale, float rbc2, float eps)
{
    float mn  = fmaf(beta1, m, om1 * g);
    float vn  = fmaf(beta2, v, om2 * (g * g));
    // v_sqrt_f32 + v_rcp_f32: ~1 ulp each, ~8 VALU/element total vs ~30
    // for the precise expansions; output error ~1e-8 relative on p.
    float den = __builtin_amdgcn_sqrtf(vn * rbc2) + eps;
    return fmaf(-(scale * mn), __builtin_amdgcn_rcpf(den), p);
}

__global__ void __launch_bounds__(THREADS)
adam_patch_async(const float* __restrict__ p, const float* __restrict__ g,
                 const float* __restrict__ m, const float* __restrict__ v,
                 float* __restrict__ out,
                 const float* __restrict__ ws, long nTiles)
{
    __shared__ v4f stage[2][4][TILE_V4];   // 32 KB of the 320 KB/WGP LDS

    const int tid = threadIdx.x;

    // Tiny uniform prologue: 7 scalar loads, zero transcendentals.
    const float beta1 = ws[0], om1 = ws[1];
    const float beta2 = ws[2], om2 = ws[3];
    const float scale = ws[4], rbc2 = ws[5], eps = ws[6];

    const v4f* p4 = (const v4f*)p;
    const v4f* g4 = (const v4f*)g;
    const v4f* m4 = (const v4f*)m;
    const v4f* w4 = (const v4f*)v;
    v4f*       o4 = (v4f*)out;

    long       t      = blockIdx.x;           // tile index (uniform per block)
    const long stride = (long)gridDim.x;
    int        b      = 0;

    if (t < nTiles)
        async_tile_load(stage[0], p4, g4, m4, w4, t * TILE_V4, tid);

    while (t < nTiles) {
        long tn = t + stride;                 // uniform: nTiles is block-granular
        if (tn < nTiles) {
            async_tile_load(stage[b ^ 1], p4, g4, m4, w4, tn * TILE_V4, tid);
            // 4 newer ops outstanding; in-order completion => tile t resident
            asm volatile("s_wait_asynccnt 4" ::: "memory");
        } else {
            asm volatile("s_wait_asynccnt 0" ::: "memory");
        }

        v4f P = stage[b][0][tid];             // ds_load_b128
        v4f G = stage[b][1][tid];
        v4f M = stage[b][2][tid];
        v4f V = stage[b][3][tid];

        v4f R;
        R.x = adam1(P.x, G.x, M.x, V.x, beta1, om1, beta2, om2, scale, rbc2, eps);
        R.y = adam1(P.y, G.y, M.y, V.y, beta1, om1, beta2, om2, scale, rbc2, eps);
        R.z = adam1(P.z, G.z, M.z, V.z, beta1, om1, beta2, om2, scale, rbc2, eps);
        R.w = adam1(P.w, G.w, M.w, V.w, beta1, om1, beta2, om2, scale, rbc2, eps);

        __builtin_nontemporal_store(R, &o4[t * TILE_V4 + tid]);  // streaming store

        t = tn;
        b ^= 1;
    }
}

// Generic tail (never launched for these shapes: all sizes % 1024 == 0)
__global__ void __launch_bounds__(THREADS)
adam_patch_tail(const float* __restrict__ p, const float* __restrict__ g,
                const float* __restrict__ m, const float* __restrict__ v,
                float* __restrict__ out, long start, long n,
                const float* __restrict__ ws)
{
    long i = start + (long)blockIdx.x * blockDim.x + threadIdx.x;
    if (i >= n) return;
    out[i] = adam1(p[i], g[i], m[i], v[i],
                   ws[0], ws[1], ws[2], ws[3], ws[4], ws[5], ws[6]);
}

// ------------------------------------------------------------------ host ----
extern "C" void kernel_launch(void* const* d_in, const int* in_sizes, int n_in,
                              void* d_out, int out_size, void* d_ws, size_t ws_size,
                              hipStream_t stream)
{
    const float* s_alpha = (const float*)d_in[0];
    const float* s_b1    = (const float*)d_in[1];
    const float* s_b2    = (const float*)d_in[2];
    const float* s_leps  = (const float*)d_in[3];
    const int*   s_t     = (const int*)d_in[4];
    float*       out     = (float*)d_out;
    float*       ws      = (float*)d_ws;

    // One-thread scalar prep: removes per-wave tanh/pow SALU expansions
    // from the streaming kernels entirely.
    adam_setup<<<1, 1, 0, stream>>>(s_alpha, s_b1, s_b2, s_leps, s_t, ws);

    long offset = 0;
    for (int k = 0; k < 3; ++k) {
        const float* p = (const float*)d_in[5 + 4 * k + 0];
        const float* g = (const float*)d_in[5 + 4 * k + 1];
        const float* m = (const float*)d_in[5 + 4 * k + 2];
        const float* v = (const float*)d_in[5 + 4 * k + 3];
        const long   n = (long)in_sizes[5 + 4 * k + 0];

        const long nTiles = n / (TILE_V4 * 4);
        if (nTiles > 0) {
            int blocks = (int)(nTiles < 8192 ? nTiles : 8192);
            adam_patch_async<<<blocks, THREADS, 0, stream>>>(
                p, g, m, v, out + offset, ws, nTiles);
        }
        const long start = nTiles * (long)(TILE_V4 * 4);
        const long rem   = n - start;
        if (rem > 0) {
            int blocks = (int)((rem + THREADS - 1) / THREADS);
            adam_patch_tail<<<blocks, THREADS, 0, stream>>>(
                p, g, m, v, out + offset, start, n, ws);
        }
        offset += n;
    }
}